// Hawkes_process_Layer_7550552506803
// MI455X (gfx1250) — compile-verified
//
#include <hip/hip_runtime.h>
#include <hip/hip_bf16.h>

typedef __attribute__((ext_vector_type(2))) float v2f;
typedef __attribute__((ext_vector_type(8))) float v8f;

#define DIM  128
#define META 5000

// ---------------------------------------------------------------------------
// Kernel 1: zero the output accumulator (harness poisons d_out with 0xAA).
// ---------------------------------------------------------------------------
__global__ void zero_f4(float4* __restrict__ out, int n4) {
    int i      = blockIdx.x * blockDim.x + threadIdx.x;
    int stride = gridDim.x * blockDim.x;
    float4 z = make_float4(0.f, 0.f, 0.f, 0.f);
    for (; i < n4; i += stride) out[i] = z;
}

// ---------------------------------------------------------------------------
// Kernel 2: proj[m] = dot(embedding[m,:], params[:,0]) via V_WMMA_F32_16X16X4_F32.
// One wave handles a 16-row tile. B is params broadcast across all 16 N
// columns, so every column of D equals the projection for rows m_base..+15.
// A layout (32-bit A 16x4, ISA 7.12.2): lanes 0-15 hold K=0 (v0), K=1 (v1);
// lanes 16-31 hold K=2 (v0), K=3 (v1). B mirrors with N across lanes; since
// B[k][n] is constant over n, each lane only needs params[kb+koff(+1)].
// D layout: lane 0 holds D[0..7][0] in c[0..7]; lane 16 holds D[8..15][0].
// ---------------------------------------------------------------------------
__global__ void proj_wmma(const float* __restrict__ emb,
                          const float* __restrict__ params,
                          float* __restrict__ proj, int n_nodes) {
    int wave   = (blockIdx.x * blockDim.x + threadIdx.x) >> 5;
    int lane   = threadIdx.x & 31;
    int m_base = wave * 16;
    if (m_base >= n_nodes) return;          // wave-uniform: EXEC all-1 inside

    int mrow = m_base + (lane & 15);
    if (mrow >= n_nodes) mrow = n_nodes - 1;    // safe clamp for ragged tile
    int koff = (lane >> 4) * 2;                 // 0 for lanes 0-15, 2 for 16-31

    const float* __restrict__ arow = emb + (size_t)mrow * DIM;
    v8f c = {};
#pragma unroll
    for (int kb = 0; kb < DIM; kb += 4) {
        v2f a, b;
        a.x = arow[kb + koff];
        a.y = arow[kb + koff + 1];
        b.x = params[kb + koff];
        b.y = params[kb + koff + 1];
        // (neg_a, A, neg_b, B, c_mod, C, reuse_a, reuse_b)
        c = __builtin_amdgcn_wmma_f32_16x16x4_f32(
                false, a, false, b, (short)0, c, false, false);
    }

    if (lane == 0) {
#pragma unroll
        for (int r = 0; r < 8; ++r)
            if (m_base + r < n_nodes) proj[m_base + r] = c[r];
    } else if (lane == 16) {
#pragma unroll
        for (int r = 0; r < 8; ++r)
            if (m_base + 8 + r < n_nodes) proj[m_base + 8 + r] = c[r];
    }
}

// ---------------------------------------------------------------------------
// Kernel 3: fused gather / exp / scaled atomic scatter. One wave per edge:
// lane l handles dims [4l, 4l+4) via a float4 load (coalesced 512B row read,
// L2-resident) and 4 hardware global_atomic_add_f32 into the output slice.
// Rows outside [row_lo, row_hi) fall outside the returned slice -> skipped.
// ---------------------------------------------------------------------------
__global__ void hawkes_scatter(const float* __restrict__ interval,
                               const float* __restrict__ emb,
                               const long long* __restrict__ row,
                               const long long* __restrict__ col,
                               const float* __restrict__ proj,
                               float* __restrict__ out,
                               int n_edges, int row_lo, int row_hi) {
    int lane   = threadIdx.x & 31;
    int wid    = (blockIdx.x * blockDim.x + threadIdx.x) >> 5;
    int nwaves = (gridDim.x * blockDim.x) >> 5;

    for (int e = wid; e < n_edges; e += nwaves) {
        long long r = row[e];
        long long c = col[e];

        // prefetch next iteration's embedding row into cache (global_prefetch_b8)
        int en = e + nwaves;
        if (en < n_edges) {
            long long cn = col[en];
            __builtin_prefetch(emb + (size_t)cn * DIM + lane * 4, 0, 3);
        }

        float theta = proj[c];
        float decay = expf(interval[e] * theta);

        if (r >= row_lo && r < row_hi) {
            const float4* __restrict__ src = (const float4*)(emb + (size_t)c * DIM);
            float4 v = src[lane];
            float* dst = out + (size_t)(r - row_lo) * DIM + lane * 4;
            unsafeAtomicAdd(dst + 0, decay * v.x);
            unsafeAtomicAdd(dst + 1, decay * v.y);
            unsafeAtomicAdd(dst + 2, decay * v.z);
            unsafeAtomicAdd(dst + 3, decay * v.w);
        }
    }
}

// ---------------------------------------------------------------------------
extern "C" void kernel_launch(void* const* d_in, const int* in_sizes, int n_in,
                              void* d_out, int out_size, void* d_ws, size_t ws_size,
                              hipStream_t stream) {
    const float*     interval = (const float*)d_in[0];
    const float*     emb      = (const float*)d_in[1];
    const float*     params   = (const float*)d_in[2];
    const long long* eidx     = (const long long*)d_in[3];
    // d_in[4] = author_num scalar; rows_out derived from out_size instead.

    int n_edges  = in_sizes[3] / 2;
    int n_nodes  = in_sizes[1] / DIM;
    int rows_out = out_size / DIM;

    float* out  = (float*)d_out;
    float* proj = (float*)d_ws;            // n_nodes floats (~200 KB)

    // 1) zero the output
    int n4 = out_size / 4;
    int zb = (n4 + 255) / 256;
    if (zb > 4096) zb = 4096;
    hipLaunchKernelGGL(zero_f4, dim3(zb), dim3(256), 0, stream,
                       (float4*)out, n4);

    // 2) projection via f32 WMMA: one wave per 16 rows
    int tiles   = (n_nodes + 15) / 16;
    int threads = tiles * 32;
    int pb      = (threads + 255) / 256;
    hipLaunchKernelGGL(proj_wmma, dim3(pb), dim3(256), 0, stream,
                       emb, params, proj, n_nodes);

    // 3) fused gather/exp/atomic-scatter: one wave per edge, grid-stride
    hipLaunchKernelGGL(hawkes_scatter, dim3(2048), dim3(256), 0, stream,
                       interval, emb, eidx, eidx + n_edges, proj, out,
                       n_edges, META, META + rows_out);
}